// AttGNN_4277787427601
// MI455X (gfx1250) — compile-verified
//
#include <hip/hip_runtime.h>
#include <math.h>

typedef __attribute__((ext_vector_type(2))) float v2f;
typedef __attribute__((ext_vector_type(8))) float v8f;

#define F_IN  20
#define F_OUT 20
#define N_OUT 5
#define MAXB  64

__device__ __forceinline__ float leaky(float v, float s) { return v > 0.f ? v : v * s; }
// order-preserving float->uint encoding for atomicMax
__device__ __forceinline__ unsigned fenc(float f) {
    unsigned u = __float_as_uint(f);
    return (u & 0x80000000u) ? ~u : (u | 0x80000000u);
}
__device__ __forceinline__ float fdec(unsigned u) {
    return __uint_as_float((u & 0x80000000u) ? (u & 0x7fffffffu) : ~u);
}

// ---------------------------------------------------------------------------
// Pack [W | att_src | att_dst | 0-pad] into WMMA B-fragment order:
// Bfrag[((step*2 + tile)*32 + lane)*2 + {0,1}]
//   step = 0..4 (K chunks of 4), tile = 0..1 (cols 0-15 / 16-31),
//   lane<16 holds K = 4*step+{0,1}; lane>=16 holds K = 4*step+{2,3};
//   logical col = tile*16 + (lane&15); col 20 -> att_src, 21 -> att_dst.
// 640 floats total, one tiny block.
// ---------------------------------------------------------------------------
__global__ void pack_bfrag(const float* __restrict__ W, const float* __restrict__ att_src,
                           const float* __restrict__ att_dst, float* __restrict__ Bfrag)
{
    int idx = threadIdx.x;                 // 0..319
    if (idx >= 320) return;
    int kk  = idx >> 6;                    // step
    int rem = idx & 63;
    int t   = rem >> 5;                    // tile
    int l   = rem & 31;                    // lane
    int kh  = l >> 4;
    int n0  = l & 15;
    int k0  = kk * 4 + kh * 2;
    int col = t * 16 + n0;
    float v0, v1;
    if (col < F_OUT)           { v0 = W[k0 * F_OUT + col]; v1 = W[(k0 + 1) * F_OUT + col]; }
    else if (col == F_OUT)     { v0 = att_src[k0];         v1 = att_src[k0 + 1]; }
    else if (col == F_OUT + 1) { v0 = att_dst[k0];         v1 = att_dst[k0 + 1]; }
    else                       { v0 = 0.f;                 v1 = 0.f; }
    Bfrag[(size_t)idx * 2 + 0] = v0;
    Bfrag[(size_t)idx * 2 + 1] = v1;
}

// ---------------------------------------------------------------------------
// Pass A: h = x@W (WMMA f32 16x16x4); a_s/a_d fall out of tile-1 cols 20/21.
// One wave per 16-row tile; branch-free loads (clamped A rows, packed B).
// ---------------------------------------------------------------------------
__global__ void gat_gemm_wmma(const float* __restrict__ x, const float* __restrict__ Bfrag,
                              float* __restrict__ h, float* __restrict__ as_,
                              float* __restrict__ ad_, int n)
{
    int wave = blockIdx.x * (blockDim.x >> 5) + (threadIdx.x >> 5);
    int row0 = wave * 16;
    if (row0 >= n) return;
    int lane = threadIdx.x & 31;
    int m    = lane & 15;
    int kh   = lane >> 4;

    // A rows: clamp OOB rows to n-1 (their D rows are never stored)
    int arow = row0 + m;
    if (arow >= n) arow = n - 1;

    // Preload all fragments (branch-free b64 loads)
    const v2f* Bf = (const v2f*)Bfrag;
    v2f bfr[10];
#pragma unroll
    for (int q = 0; q < 10; ++q) bfr[q] = Bf[q * 32 + lane];
    v2f afr[5];
#pragma unroll
    for (int kk = 0; kk < 5; ++kk)
        afr[kk] = *(const v2f*)(x + (size_t)arow * F_IN + kk * 4 + kh * 2);

    v8f c0 = {0.f,0.f,0.f,0.f,0.f,0.f,0.f,0.f};
    v8f c1 = {0.f,0.f,0.f,0.f,0.f,0.f,0.f,0.f};
#pragma unroll
    for (int kk = 0; kk < 5; ++kk) {
        c0 = __builtin_amdgcn_wmma_f32_16x16x4_f32(false, afr[kk], false, bfr[kk*2+0],
                                                   (short)0, c0, false, false);
        c1 = __builtin_amdgcn_wmma_f32_16x16x4_f32(false, afr[kk], false, bfr[kk*2+1],
                                                   (short)0, c1, false, false);
    }

    int col = 16 + m;
#pragma unroll
    for (int r = 0; r < 8; ++r) {
        int orow = row0 + r + 8 * kh;       // C/D layout: lanes>=16 hold M=r+8
        if (orow < n) {
            h[(size_t)orow * F_OUT + m] = c0[r];
            float v = c1[r];
            if (col < F_OUT)           h[(size_t)orow * F_OUT + col] = v;
            else if (col == F_OUT)     as_[orow] = v;
            else if (col == F_OUT + 1) ad_[orow] = v;
        }
    }
}

// ---------------------------------------------------------------------------
// Pass B: per-edge logit + segment max over dst (self-loops appended)
// ---------------------------------------------------------------------------
__global__ void edge_max_kernel(const int* __restrict__ ei, int E, int n,
                                const float* __restrict__ as_, const float* __restrict__ ad_,
                                unsigned* __restrict__ mEnc)
{
    int i = blockIdx.x * blockDim.x + threadIdx.x;
    int total = E + n;
    if (i >= total) return;
    int s, d;
    if (i < E) { s = ei[i]; d = ei[E + i]; } else { s = d = i - E; }
    float e = leaky(as_[s] + ad_[d], 0.2f);
    atomicMax(&mEnc[d], fenc(e));
}

// ---------------------------------------------------------------------------
// Pass C: ex = exp(e - m[dst]); den[dst] += ex; cache ex per edge
// ---------------------------------------------------------------------------
__global__ void edge_expsum_kernel(const int* __restrict__ ei, int E, int n,
                                   const float* __restrict__ as_, const float* __restrict__ ad_,
                                   const unsigned* __restrict__ mEnc,
                                   float* __restrict__ den, float* __restrict__ exv)
{
    int i = blockIdx.x * blockDim.x + threadIdx.x;
    int total = E + n;
    if (i >= total) return;
    int s, d;
    if (i < E) { s = ei[i]; d = ei[E + i]; } else { s = d = i - E; }
    float e  = leaky(as_[s] + ad_[d], 0.2f);
    float ex = expf(e - fdec(mEnc[d]));
    exv[i] = ex;
    atomicAdd(&den[d], ex);
}

// ---------------------------------------------------------------------------
// Pass D: agg[dst] += (ex/den[dst]) * h[src]   (float4 gathers, f32 atomics)
// ---------------------------------------------------------------------------
__global__ void edge_aggregate_kernel(const int* __restrict__ ei, int E, int n,
                                      const float* __restrict__ exv, const float* __restrict__ den,
                                      const float* __restrict__ h, float* __restrict__ agg)
{
    int i = blockIdx.x * blockDim.x + threadIdx.x;
    int total = E + n;
    if (i >= total) return;
    int s, d;
    if (i < E) { s = ei[i]; d = ei[E + i]; } else { s = d = i - E; }
    float alpha = exv[i] / den[d];
    const float4* hr = (const float4*)(h + (size_t)s * F_OUT);   // 80B row stride, 16B aligned
    float* ar = agg + (size_t)d * F_OUT;
#pragma unroll
    for (int q = 0; q < 5; ++q) {
        float4 v = hr[q];
        atomicAdd(ar + q * 4 + 0, alpha * v.x);
        atomicAdd(ar + q * 4 + 1, alpha * v.y);
        atomicAdd(ar + q * 4 + 2, alpha * v.z);
        atomicAdd(ar + q * 4 + 3, alpha * v.w);
    }
}

// ---------------------------------------------------------------------------
// Pass E: xact = LeakyReLU(agg + bias, 0.01); per-graph sum + count,
// pre-reduced per block in LDS (batch sorted -> few graphs per block)
// ---------------------------------------------------------------------------
__global__ void relu_pool_kernel(const float* __restrict__ agg, const float* __restrict__ bias,
                                 const int* __restrict__ batch, int n,
                                 float* __restrict__ pooled, float* __restrict__ cnt)
{
    __shared__ float sPool[MAXB * F_OUT];
    __shared__ float sCnt[MAXB];
    for (int t = threadIdx.x; t < MAXB * F_OUT; t += blockDim.x) sPool[t] = 0.f;
    for (int t = threadIdx.x; t < MAXB; t += blockDim.x) sCnt[t] = 0.f;
    __syncthreads();

    int i = blockIdx.x * blockDim.x + threadIdx.x;
    if (i < n) {
        int b = batch[i];
        atomicAdd(&sCnt[b], 1.f);
        const float4* ar = (const float4*)(agg + (size_t)i * F_OUT);
        const float4* br = (const float4*)bias;
#pragma unroll
        for (int q = 0; q < 5; ++q) {
            float4 v = ar[q];
            float4 bb = br[q];
            atomicAdd(&sPool[b * F_OUT + q * 4 + 0], leaky(v.x + bb.x, 0.01f));
            atomicAdd(&sPool[b * F_OUT + q * 4 + 1], leaky(v.y + bb.y, 0.01f));
            atomicAdd(&sPool[b * F_OUT + q * 4 + 2], leaky(v.z + bb.z, 0.01f));
            atomicAdd(&sPool[b * F_OUT + q * 4 + 3], leaky(v.w + bb.w, 0.01f));
        }
    }
    __syncthreads();
    for (int t = threadIdx.x; t < MAXB * F_OUT; t += blockDim.x) {
        int g = t / F_OUT;
        if (sCnt[g] > 0.f) atomicAdd(&pooled[t], sPool[t]);
    }
    for (int t = threadIdx.x; t < MAXB; t += blockDim.x)
        if (sCnt[t] > 0.f) atomicAdd(&cnt[t], sCnt[t]);
}

// ---------------------------------------------------------------------------
// Pass F: logits = (pooled/cnt) @ out_W + out_b ; softmax  (B=64, tiny)
// ---------------------------------------------------------------------------
__global__ void head_kernel(const float* __restrict__ pooled, const float* __restrict__ cnt,
                            const float* __restrict__ out_W, const float* __restrict__ out_b,
                            float* __restrict__ out, int nB)
{
    int g = blockIdx.x * blockDim.x + threadIdx.x;
    if (g >= nB) return;
    float inv = 1.f / fmaxf(cnt[g], 1.f);
    float logits[N_OUT];
#pragma unroll
    for (int j = 0; j < N_OUT; ++j) logits[j] = out_b[j];
    for (int f = 0; f < F_OUT; ++f) {
        float p = pooled[g * F_OUT + f] * inv;
#pragma unroll
        for (int j = 0; j < N_OUT; ++j) logits[j] += p * out_W[f * N_OUT + j];
    }
    float mx = logits[0];
#pragma unroll
    for (int j = 1; j < N_OUT; ++j) mx = fmaxf(mx, logits[j]);
    float ev[N_OUT], sum = 0.f;
#pragma unroll
    for (int j = 0; j < N_OUT; ++j) { ev[j] = expf(logits[j] - mx); sum += ev[j]; }
#pragma unroll
    for (int j = 0; j < N_OUT; ++j) out[g * N_OUT + j] = ev[j] / sum;
}

// ---------------------------------------------------------------------------
extern "C" void kernel_launch(void* const* d_in, const int* in_sizes, int n_in,
                              void* d_out, int out_size, void* d_ws, size_t ws_size,
                              hipStream_t stream)
{
    const float* x       = (const float*)d_in[0];
    const int*   ei      = (const int*)d_in[1];
    const int*   batch   = (const int*)d_in[2];
    const float* W       = (const float*)d_in[3];
    const float* att_src = (const float*)d_in[4];
    const float* att_dst = (const float*)d_in[5];
    const float* bias    = (const float*)d_in[6];
    const float* out_W   = (const float*)d_in[7];
    const float* out_b   = (const float*)d_in[8];
    float* out = (float*)d_out;

    int n  = in_sizes[0] / F_IN;
    int E  = in_sizes[1] / 2;
    int nB = out_size / N_OUT;
    int total = E + n;

    // workspace carve (256B aligned)
    size_t off = 0;
    auto carve = [&](size_t bytes) -> char* {
        char* p = (char*)d_ws + off;
        off += (bytes + 255) & ~(size_t)255;
        return p;
    };
    float*    h      = (float*)carve((size_t)n * F_OUT * 4);
    float*    exv    = (float*)carve((size_t)total * 4);
    float*    as_    = (float*)carve((size_t)n * 4);
    float*    ad_    = (float*)carve((size_t)n * 4);
    float*    Bfrag  = (float*)carve(640 * 4);
    char* zbase = (char*)d_ws + off;                      // zero-init region starts here
    float*    agg    = (float*)carve((size_t)n * F_OUT * 4);
    unsigned* mEnc   = (unsigned*)carve((size_t)n * 4);   // enc=0 < enc of any real float
    float*    den    = (float*)carve((size_t)n * 4);
    float*    pooled = (float*)carve((size_t)nB * F_OUT * 4);
    float*    cnt    = (float*)carve((size_t)nB * 4);
    size_t zbytes = ((char*)d_ws + off) - zbase;
    hipMemsetAsync(zbase, 0, zbytes, stream);

    // Pass A: pack B fragments, then WMMA node transform
    pack_bfrag<<<1, 320, 0, stream>>>(W, att_src, att_dst, Bfrag);
    {
        int tiles  = (n + 15) / 16;
        int blocks = (tiles + 7) / 8;   // 8 waves per 256-thread block
        gat_gemm_wmma<<<blocks, 256, 0, stream>>>(x, Bfrag, h, as_, ad_, n);
    }
    int eblocks = (total + 255) / 256;
    edge_max_kernel      <<<eblocks, 256, 0, stream>>>(ei, E, n, as_, ad_, mEnc);
    edge_expsum_kernel   <<<eblocks, 256, 0, stream>>>(ei, E, n, as_, ad_, mEnc, den, exv);
    edge_aggregate_kernel<<<eblocks, 256, 0, stream>>>(ei, E, n, exv, den, h, agg);
    relu_pool_kernel     <<<(n + 255) / 256, 256, 0, stream>>>(agg, bias, batch, n, pooled, cnt);
    head_kernel          <<<(nB + 63) / 64, 64, 0, stream>>>(pooled, cnt, out_W, out_b, out, nB);
}